// MomentFlowMambaCell_54820962566290
// MI455X (gfx1250) — compile-verified
//
#include <hip/hip_runtime.h>
#include <hip/hip_bf16.h>

typedef float v2f __attribute__((ext_vector_type(2)));
typedef float v8f __attribute__((ext_vector_type(8)));

#define HW 4096          // 64*64
#define PITCH 66         // LDS row pitch for m0 (conflict-free b64 frag loads)
#define TWO_PI 6.28318530717958647692f

// ---------------------------------------------------------------------------
// Kernel 1: spectral differentiation kernel d[m] = Re(IDFT(i*k))[m]
//   d[m] = -(1/64) * sum_j k_j * sin(2*pi*(j*m mod 64)/64),  k_j = 2*pi*fftfreq
// ---------------------------------------------------------------------------
__global__ void diffkern_kernel(float* __restrict__ dk) {
    int m = threadIdx.x;
    if (m >= 64) return;
    float sum = 0.f;
    for (int j = 0; j < 64; ++j) {
        int f = (j < 32) ? j : j - 64;
        float k = TWO_PI * (float)f * (1.0f / 64.0f);
        int r = (j * m) & 63;
        sum -= k * sinf(TWO_PI * (float)r * (1.0f / 64.0f));
    }
    dk[m] = sum * (1.0f / 64.0f);
}

// ---------------------------------------------------------------------------
// Kernel 2: group-norm statistics. One block per (b,g); 16 ch * 4096 px.
// ---------------------------------------------------------------------------
__global__ __launch_bounds__(256) void gn_stats_kernel(const float* __restrict__ u,
                                                       float* __restrict__ stats) {
    const int bg = blockIdx.x;                      // b*4 + g
    const float* p = u + (size_t)bg * 65536;        // channels of a group are contiguous
    float s = 0.f, ss = 0.f;
    for (int i = threadIdx.x; i < 65536; i += 256) {
        float v = p[i];
        s += v; ss += v * v;
    }
    __shared__ float r1[256], r2[256];
    r1[threadIdx.x] = s; r2[threadIdx.x] = ss;
    __syncthreads();
    for (int off = 128; off > 0; off >>= 1) {
        if (threadIdx.x < off) {
            r1[threadIdx.x] += r1[threadIdx.x + off];
            r2[threadIdx.x] += r2[threadIdx.x + off];
        }
        __syncthreads();
    }
    if (threadIdx.x == 0) {
        float mu  = r1[0] * (1.0f / 65536.0f);
        float var = r2[0] * (1.0f / 65536.0f) - mu * mu;
        stats[bg * 2 + 0] = mu;
        stats[bg * 2 + 1] = rsqrtf(var + 1e-5f);
    }
}

// ---------------------------------------------------------------------------
// Kernel 3: normalize (float4 elementwise). 262144 float4s.
// ---------------------------------------------------------------------------
__global__ __launch_bounds__(256) void normalize_kernel(const float* __restrict__ u,
                                                        const float* __restrict__ stats,
                                                        const float* __restrict__ gamma,
                                                        const float* __restrict__ beta,
                                                        float* __restrict__ un) {
    int idx4 = blockIdx.x * 256 + threadIdx.x;
    if (idx4 >= 262144) return;
    int bc = idx4 >> 10;                // b*64 + c   (1024 float4 per plane)
    int c  = bc & 63;
    int bg = bc >> 4;                   // b*4 + c/16
    float mu = stats[bg * 2 + 0];
    float rs = stats[bg * 2 + 1];
    float ga = gamma[c], be = beta[c];
    float4 v = ((const float4*)u)[idx4];
    v.x = (v.x - mu) * rs * ga + be;
    v.y = (v.y - mu) * rs * ga + be;
    v.z = (v.z - mu) * rs * ga + be;
    v.w = (v.w - mu) * rs * ga + be;
    ((float4*)un)[idx4] = v;
}

// ---------------------------------------------------------------------------
// Kernel 4: 3x3 circular convs -> delta (softplus+clip), B_val, C_val.
// grid = (16 pixel tiles, 6 out-channel groups, 4 batches), 256 threads.
// Input channels processed in 2 chunks of 32 to bound LDS (32*18*18 f32).
// ---------------------------------------------------------------------------
__global__ __launch_bounds__(256) void conv_kernel(const float* __restrict__ un,
                                                   const float* __restrict__ w_delta,
                                                   const float* __restrict__ b_delta,
                                                   const float* __restrict__ w_B,
                                                   const float* __restrict__ w_C,
                                                   const float* __restrict__ dt_inv,
                                                   float* __restrict__ delta_o,
                                                   float* __restrict__ Bv_o,
                                                   float* __restrict__ Cv_o) {
    __shared__ float s_un[32 * 324];    // 32 channels x 18x18 halo tile (41.5 KB)

    const int tile = blockIdx.x;        // 0..15
    const int cg   = blockIdx.y;        // 0..5 : 0-3 delta, 4 B_val, 5 C_val
    const int b    = blockIdx.z;        // 0..3
    const int ty0 = (tile >> 2) * 16, tx0 = (tile & 3) * 16;
    const int tid = threadIdx.x;
    const int py = tid >> 4, px = tid & 15;

    const float* ub = un + ((size_t)b << 18);

    const float* wb;
    int ocbase;
    if (cg < 4)      { wb = w_delta + (size_t)(cg * 16) * 576; ocbase = cg * 16; }
    else if (cg == 4){ wb = w_B; ocbase = 0; }
    else             { wb = w_C; ocbase = 0; }
    const float dti = dt_inv[0];

    float acc[16];
#pragma unroll
    for (int oc = 0; oc < 16; ++oc) acc[oc] = 0.f;

    for (int cc = 0; cc < 64; cc += 32) {
        __syncthreads();
        for (int i = tid; i < 32 * 324; i += 256) {
            int c = i / 324, r = i - c * 324;
            int yy = r / 18, xx = r - yy * 18;
            int gy = (ty0 + yy - 1) & 63;
            int gx = (tx0 + xx - 1) & 63;
            s_un[i] = ub[(((size_t)(cc + c)) << 12) + (gy << 6) + gx];
        }
        __syncthreads();

        for (int oc = 0; oc < 16; ++oc) {
            const float* wp = wb + (size_t)oc * 576 + (size_t)cc * 9;
            float a = acc[oc];
            for (int c = 0; c < 32; ++c) {
                const float* t  = &s_un[c * 324 + py * 18 + px];
                const float* w9 = wp + c * 9;                 // wave-uniform -> s_load
                a = fmaf(t[0],  w9[0], a);
                a = fmaf(t[1],  w9[1], a);
                a = fmaf(t[2],  w9[2], a);
                a = fmaf(t[18], w9[3], a);
                a = fmaf(t[19], w9[4], a);
                a = fmaf(t[20], w9[5], a);
                a = fmaf(t[36], w9[6], a);
                a = fmaf(t[37], w9[7], a);
                a = fmaf(t[38], w9[8], a);
            }
            acc[oc] = a;
        }
    }

    const int h = ty0 + py, w = tx0 + px;
#pragma unroll
    for (int oc = 0; oc < 16; ++oc) {
        int o = ocbase + oc;
        if (cg < 4) {
            float x  = acc[oc] + b_delta[o] + dti;
            float sp = (x > 20.f) ? x : log1pf(expf(x));      // stable softplus
            sp = fminf(fmaxf(sp, 1e-4f), 5.0f);
            delta_o[(((size_t)(b * 64 + o)) << 12) + (h << 6) + w] = sp;
        } else if (cg == 4) {
            Bv_o[(((size_t)(b * 16 + o)) << 12) + (h << 6) + w] = acc[oc];
        } else {
            Cv_o[(((size_t)(b * 16 + o)) << 12) + (h << 6) + w] = acc[oc];
        }
    }
}

// ---------------------------------------------------------------------------
// Kernel 5: fused state update. One block (256 thr = 8 waves) per (b,d).
// Per n: stage m0 in LDS, compute grad_x = M0 @ Dm and grad_y = Dm^T @ M0
// with V_WMMA_F32_16X16X4_F32 (Dm[k,j] = d[(j-k)&63] generated from s_dk),
// then fused elementwise state update + y-accumulation on fragment pixels.
// ---------------------------------------------------------------------------
__global__ __launch_bounds__(256) void state_kernel(const float* __restrict__ u,
                                                    const float* __restrict__ s_prev,
                                                    const float* __restrict__ log_A_real,
                                                    const float* __restrict__ D_param,
                                                    const float* __restrict__ dk,
                                                    const float* __restrict__ delta_ws,
                                                    const float* __restrict__ Bval,
                                                    const float* __restrict__ Cval,
                                                    float* __restrict__ y_out,
                                                    float* __restrict__ s_out) {
    __shared__ float s_dk[64];
    __shared__ float s_m0[64 * PITCH];  // 16.9 KB, pitch 66 -> conflict-free b64 loads
    __shared__ float s_delta[HW];       // 16 KB
    __shared__ float s_y[HW];           // 16 KB

    const int blk = blockIdx.x;         // 0..255
    const int b = blk >> 6, d = blk & 63;
    const int tid = threadIdx.x;
    const int lane = tid & 31, wv = tid >> 5;

    if (tid < 64) s_dk[tid] = dk[tid];
    const float* up = u        + (((size_t)(b * 64 + d)) << 12);
    const float* dp = delta_ws + (((size_t)(b * 64 + d)) << 12);
    for (int i = tid; i < HW; i += 256) {
        s_delta[i] = dp[i];
        s_y[i] = 0.f;
    }
    __syncthreads();

    for (int n = 0; n < 16; ++n) {
        const size_t p0 = ((((size_t)(b * 3 + 0) * 64 + d) * 16 + n) << 12);
        const size_t p1 = ((((size_t)(b * 3 + 1) * 64 + d) * 16 + n) << 12);
        const size_t p2 = ((((size_t)(b * 3 + 2) * 64 + d) * 16 + n) << 12);
        const float* m0p = s_prev + p0;
        const float* mxp = s_prev + p1;
        const float* myp = s_prev + p2;
        float* o0 = s_out + p0;
        float* o1 = s_out + p1;
        float* o2 = s_out + p2;

        // stage m0 (64x64) into pitched LDS
        for (int j = tid; j < 1024; j += 256) {
            float4 v = ((const float4*)m0p)[j];
            int row = j >> 4, col = (j & 15) << 2;
            float* q = &s_m0[row * PITCH + col];
            q[0] = v.x; q[1] = v.y; q[2] = v.z; q[3] = v.w;
        }
        __syncthreads();

        const float Adn = -expf(log_A_real[d * 16 + n]);   // A = -exp(logA)
        const float* Bp = Bval + (((size_t)(b * 16 + n)) << 12);
        const float* Cp = Cval + (((size_t)(b * 16 + n)) << 12);

        for (int half = 0; half < 2; ++half) {
            const int t  = wv * 2 + half;          // 16 tiles of 16x16, 2 per wave
            const int tr = t >> 2, tc = t & 3;
            const int m    = lane & 15;
            const int ksel = (lane >> 4) << 1;     // lanes 16-31 hold K+2,K+3
            const int h0 = tr * 16 + m;            // A-row (grad_x) / Dm col (grad_y)
            const int w0 = tc * 16 + m;            // B-col for both GEMMs

            v8f accX = {0.f,0.f,0.f,0.f,0.f,0.f,0.f,0.f};
            v8f accY = {0.f,0.f,0.f,0.f,0.f,0.f,0.f,0.f};
#pragma unroll
            for (int kk = 0; kk < 64; kk += 4) {
                const int kb = kk + ksel;
                // grad_x: A[m,k] = m0[h0][k], B[k,nn] = Dm[k][w0] = d[(w0-k)&63]
                v2f aX = *(const v2f*)&s_m0[h0 * PITCH + kb];
                v2f bX;
                bX.x = s_dk[(w0 - kb)     & 63];
                bX.y = s_dk[(w0 - kb - 1) & 63];
                accX = __builtin_amdgcn_wmma_f32_16x16x4_f32(
                           false, aX, false, bX, (short)0, accX, false, false);
                // grad_y: A[m,k] = Dm[k][h0] = d[(h0-k)&63], B[k,nn] = m0[k][w0]
                v2f aY;
                aY.x = s_dk[(h0 - kb)     & 63];
                aY.y = s_dk[(h0 - kb - 1) & 63];
                v2f bY;
                bY.x = s_m0[kb * PITCH + w0];
                bY.y = s_m0[(kb + 1) * PITCH + w0];
                accY = __builtin_amdgcn_wmma_f32_16x16x4_f32(
                           false, aY, false, bY, (short)0, accY, false, false);
            }

            // fused elementwise state update on this wave's fragment pixels:
            // C/D layout: lane -> column, VGPR vg -> row vg (lanes<16) / vg+8
            const int wcol  = tc * 16 + (lane & 15);
            const int hbase = tr * 16 + ((lane >> 4) << 3);
#pragma unroll
            for (int vg = 0; vg < 8; ++vg) {
                const int h   = hbase + vg;
                const int pix = (h << 6) + wcol;
                float gx  = accX[vg];
                float gy  = accY[vg];
                float dv  = s_delta[pix];
                float Ab  = expf(dv * Adn);                 // A_bar
                float m0v = s_m0[h * PITCH + wcol];
                float uv  = up[pix];
                float Bv  = Bp[pix];
                float Cv  = Cp[pix];
                float mxv = mxp[pix];
                float myv = myp[pix];
                float m0n = fmaf(Ab, m0v, dv * Bv * uv);    // A_bar*m0 + B_bar*u
                o0[pix] = m0n;
                o1[pix] = Ab * (mxv - gx);
                o2[pix] = Ab * (myv - gy);
                s_y[pix] += m0n * Cv;                       // unique owner per pixel
            }
        }
        __syncthreads();
    }

    // y_new = y_state + u * D_param
    const float Dp = D_param[d];
    float* yp = y_out + (((size_t)(b * 64 + d)) << 12);
    for (int i = tid; i < HW; i += 256)
        yp[i] = fmaf(up[i], Dp, s_y[i]);
}

// ---------------------------------------------------------------------------
// Launch
// ---------------------------------------------------------------------------
extern "C" void kernel_launch(void* const* d_in, const int* in_sizes, int n_in,
                              void* d_out, int out_size, void* d_ws, size_t ws_size,
                              hipStream_t stream) {
    const float* u_t        = (const float*)d_in[0];
    const float* s_prev     = (const float*)d_in[1];
    const float* gamma      = (const float*)d_in[2];
    const float* beta       = (const float*)d_in[3];
    const float* w_delta    = (const float*)d_in[4];
    const float* b_delta    = (const float*)d_in[5];
    const float* w_B        = (const float*)d_in[6];
    const float* w_C        = (const float*)d_in[7];
    const float* log_A_real = (const float*)d_in[8];
    const float* D_param    = (const float*)d_in[9];
    const float* dt_inv     = (const float*)d_in[10];

    float* y_out = (float*)d_out;                 // (4,1,64,64,64)
    float* s_out = (float*)d_out + 1048576;       // (4,3,64,16,64,64)

    // workspace layout (floats)
    float* ws    = (float*)d_ws;
    float* dk    = ws;                            // 64
    float* stats = ws + 64;                       // 32
    float* un    = ws + 128;                      // 1048576
    float* delta = un + 1048576;                  // 1048576
    float* Bv    = delta + 1048576;               // 262144
    float* Cv    = Bv + 262144;                   // 262144  (total ~10.5 MB)

    diffkern_kernel<<<1, 64, 0, stream>>>(dk);
    gn_stats_kernel<<<16, 256, 0, stream>>>(u_t, stats);
    normalize_kernel<<<1024, 256, 0, stream>>>(u_t, stats, gamma, beta, un);
    conv_kernel<<<dim3(16, 6, 4), 256, 0, stream>>>(un, w_delta, b_delta, w_B, w_C,
                                                    dt_inv, delta, Bv, Cv);
    state_kernel<<<256, 256, 0, stream>>>(u_t, s_prev, log_A_real, D_param,
                                          dk, delta, Bv, Cv, y_out, s_out);
}